// VInputConv_721554506128
// MI455X (gfx1250) — compile-verified
//
#include <hip/hip_runtime.h>

// ---------------------------------------------------------------------------
// y[b,k,ho,wo,f] = g[b,k,ho,wo] * w[k,f]        (pure rank-1 outer product)
//   g[b,k,ho,wo] = x[b, ho+kh, wo+kw, c],  k = kh*48 + kw*16 + c   (k<144)
//   g[b,144,*,*] = 1.0                                              (bias row)
// Output: 267 MB fp32 stores -> store-BW bound (~11.5us @ 23.3 TB/s).
// Strategy: V_WMMA_F32_16X16X4_F32 as a rank-1 engine; M = 16 consecutive
// flattened output elements, N = f (lands on lanes -> coalesced stores),
// K-slots 0/1 carry the (at most) two distinct weight rows a tile touches.
// ---------------------------------------------------------------------------

typedef __attribute__((ext_vector_type(2))) float v2f;
typedef __attribute__((ext_vector_type(8))) float v8f;

#define BB   16
#define HH   32
#define WW   32
#define CC   16
#define FF   32
#define HOo  30
#define WOo  30
#define KP1  145                 // K+1 = 3*3*16 + 1
#define LLe  900                 // HO*WO
#define RTOT (BB * KP1)          // 2320 flattened (b,k) rows
#define ETOT (RTOT * LLe)        // 2,088,000 output elements (x F)
#define NTILES (ETOT / 16)       // 130,500 -- exact, no remainder

__global__ __launch_bounds__(256) void outer_wmma_kernel(
    const float* __restrict__ x,     // [B,H,W,C]
    const float* __restrict__ w,     // [K+1, 1, F]
    float* __restrict__ out)         // [B,K+1,HO,WO,F] = [E, F]
{
    const unsigned lane = threadIdx.x & 31u;
    const unsigned tile = blockIdx.x * 8u + (threadIdx.x >> 5);
    if (tile >= NTILES) return;                 // wave-uniform: EXEC stays all-1s

    const unsigned e0 = tile * 16u;             // first flattened element of tile
    const unsigned r0 = e0 / LLe;               // first (b,k) row in tile
    const unsigned l15 = lane & 15u;

    // ---- per-lane gather of g for element e0 + (lane&15) --------------------
    const unsigned e = e0 + l15;
    const unsigned r = e / LLe;                 // (b,k) row of this element
    const unsigned l = e - r * LLe;             // position in HO*WO
    const unsigned b = r / KP1;
    const unsigned k = r - b * KP1;

    float g;
    if (k == 144u) {
        g = 1.0f;                               // bias row (appended ones)
    } else {
        const unsigned kh = k / 48u;
        const unsigned t1 = k - kh * 48u;
        const unsigned kw = t1 >> 4;
        const unsigned c  = t1 & 15u;
        const unsigned ho = l / 30u;
        const unsigned wo = l - ho * 30u;
        g = x[((b * HH + ho + kh) * WW + (wo + kw)) * CC + c];
    }
    const unsigned slot = r - r0;               // 0 or 1 (tile spans <=2 rows)

    // A-matrix (16x4 f32): lanes 0-15 hold K=0,1; lanes 16-31 hold K=2,3 (zeroed)
    v2f a;
    a.x = (lane < 16u && slot == 0u) ? g : 0.0f;
    a.y = (lane < 16u && slot == 1u) ? g : 0.0f;

    // B-matrix (4x16 f32): v.x = row K=0 (lanes 0-15) / row K=2 (lanes 16-31, zero)
    //                      v.y = row K=1 (lanes 0-15) / row K=3 (lanes 16-31, zero)
    const unsigned r1 = (r0 + 1u < RTOT) ? (r0 + 1u) : (RTOT - 1u);
    const unsigned k0 = r0 % KP1;
    const unsigned k1 = r1 % KP1;

    v2f bLo, bHi;                               // f 0-15 and f 16-31 halves
    bLo.x = (lane < 16u) ? w[k0 * FF + l15]        : 0.0f;
    bLo.y = (lane < 16u) ? w[k1 * FF + l15]        : 0.0f;
    bHi.x = (lane < 16u) ? w[k0 * FF + 16u + l15]  : 0.0f;
    bHi.y = (lane < 16u) ? w[k1 * FF + 16u + l15]  : 0.0f;

    // ---- rank-1 WMMA: D[m,n] = g[m] * w[k(m), n] ----------------------------
    v8f c0 = {};
    v8f c1 = {};
    c0 = __builtin_amdgcn_wmma_f32_16x16x4_f32(false, a, false, bLo,
                                               (short)0, c0, false, false);
    c1 = __builtin_amdgcn_wmma_f32_16x16x4_f32(false, a, false, bHi,
                                               (short)0, c1, false, false);

    // ---- store D: VGPR j -> elem e0+j (lanes 0-15) / e0+j+8 (lanes 16-31),
    //      N = f on lanes -> each store is 2 contiguous 64B segments ----------
    const unsigned mAdd = (lane >> 4) * 8u;
    float* o = out + (size_t)(e0 + mAdd) * FF + l15;
#pragma unroll
    for (int j = 0; j < 8; ++j) {
        o[(size_t)j * FF]      = c0[j];
        o[(size_t)j * FF + 16] = c1[j];
    }
}

extern "C" void kernel_launch(void* const* d_in, const int* in_sizes, int n_in,
                              void* d_out, int out_size, void* d_ws, size_t ws_size,
                              hipStream_t stream) {
    const float* x = (const float*)d_in[0];          // (16,32,32,16) fp32
    const float* w = (const float*)d_in[1];          // (145,1,32) fp32
    float* out = (float*)d_out;                      // (16,145,30,30,32) fp32

    const unsigned wavesPerBlock = 8;                // 256 threads = 8 wave32
    const unsigned blocks = (NTILES + wavesPerBlock - 1) / wavesPerBlock;
    outer_wmma_kernel<<<dim3(blocks), dim3(256), 0, stream>>>(x, w, out);
}